// CreateSpatialTensor_22239340659537
// MI455X (gfx1250) — compile-verified
//
#include <hip/hip_runtime.h>

#define NIDX 4096
#define OW   256
#define CT   8          // channels per block
#define BLK  256        // 8 wave32 waves

__global__ __launch_bounds__(BLK)
void CreateSpatialTensor_gather_bilinear(const float* __restrict__ f0,
                                         const float* __restrict__ f1,
                                         const float* __restrict__ f2,
                                         const float* __restrict__ f3,
                                         const int*   __restrict__ indices,
                                         float*       __restrict__ out)
{
    __shared__ int s_idx[BLK];

    const int j = blockIdx.x * BLK + threadIdx.x;

    // --- CDNA5 async global->LDS stage of this block's index slice ---------
    // Tracked with ASYNCcnt; waited on right after issue, before the level-decode below.
    {
        // Low 32 bits of the generic (flat) shared address == wave-relative
        // LDS byte offset, which is exactly what the async op's VDST wants.
        unsigned lds_off = (unsigned)(size_t)(&s_idx[threadIdx.x]);
        const int* gp = indices + j;
        asm volatile("global_load_async_to_lds_b32 %0, %1, off"
                     :: "v"(lds_off), "v"((unsigned long long)(size_t)gp)
                     : "memory");
        asm volatile("s_wait_asynccnt 0" ::: "memory");
    }
    __syncthreads();

    const int idx = s_idx[threadIdx.x];
    const int y = idx >> 8;          // OW == 256
    const int x = idx & (OW - 1);

    // --- decode (level, channel tile) from blockIdx.y ----------------------
    // tiles: L0: 64/8=8 (0..7), L1: 128/8=16 (8..23),
    //        L2: 256/8=32 (24..55), L3: 512/8=64 (56..119)
    const int t = blockIdx.y;
    const float* f; int C, H, W, cbase, c0;
    if (t < 8)       { f = f0; C = 64;  H = 256; W = 256; cbase = 0;   c0 = t        * CT; }
    else if (t < 24) { f = f1; C = 128; H = 128; W = 128; cbase = 64;  c0 = (t - 8)  * CT; }
    else if (t < 56) { f = f2; C = 256; H = 64;  W = 64;  cbase = 192; c0 = (t - 24) * CT; }
    else             { f = f3; C = 512; H = 32;  W = 32;  cbase = 448; c0 = (t - 56) * CT; }

    const int b = blockIdx.z;

    // --- bilinear setup, half-pixel centers (align_corners=False) ----------
    const float scale = (float)H * (1.0f / 256.0f);   // 1, .5, .25, .125
    const float sy = fmaf((float)y + 0.5f, scale, -0.5f);
    const float sx = fmaf((float)x + 0.5f, scale, -0.5f);
    const float fy = floorf(sy), fx = floorf(sx);
    const float ty = sy - fy,    tx = sx - fx;
    const int iy0 = (int)fy, ix0 = (int)fx;
    const int iy0c = min(max(iy0,     0), H - 1);
    const int iy1c = min(max(iy0 + 1, 0), H - 1);
    const int ix0c = min(max(ix0,     0), W - 1);
    const int ix1c = min(max(ix0 + 1, 0), W - 1);

    const float w00 = (1.0f - ty) * (1.0f - tx);
    const float w01 = (1.0f - ty) * tx;
    const float w10 = ty * (1.0f - tx);
    const float w11 = ty * tx;

    const int o00 = iy0c * W + ix0c;
    const int o01 = iy0c * W + ix1c;
    const int o10 = iy1c * W + ix0c;
    const int o11 = iy1c * W + ix1c;

    const size_t HW = (size_t)H * (size_t)W;
    const float* plane = f + ((size_t)b * (size_t)C + (size_t)c0) * HW;
    float* op = out + ((size_t)b * 960 + (size_t)(cbase + c0)) * (size_t)NIDX + (size_t)j;

    // 8 channels per thread: 32 independent L2-resident gathers in flight.
    // Output is write-once -> non-temporal stores keep the 120MB pyramid
    // resident in the 192MB L2 instead of being evicted by the 63MB output.
#pragma unroll
    for (int cc = 0; cc < CT; ++cc) {
        const float v00 = plane[o00];
        const float v01 = plane[o01];
        const float v10 = plane[o10];
        const float v11 = plane[o11];
        const float v = fmaf(w00, v00, fmaf(w01, v01, fmaf(w10, v10, w11 * v11)));
        __builtin_nontemporal_store(v, op + (size_t)cc * NIDX);
        plane += HW;
    }
}

extern "C" void kernel_launch(void* const* d_in, const int* in_sizes, int n_in,
                              void* d_out, int out_size, void* d_ws, size_t ws_size,
                              hipStream_t stream) {
    const float* f0      = (const float*)d_in[0];
    const float* f1      = (const float*)d_in[1];
    const float* f2      = (const float*)d_in[2];
    const float* f3      = (const float*)d_in[3];
    const int*   indices = (const int*)  d_in[4];
    float*       out     = (float*)d_out;

    dim3 grid(NIDX / BLK, /*channel tiles*/ 120, /*batch*/ 4);
    dim3 block(BLK, 1, 1);
    CreateSpatialTensor_gather_bilinear<<<grid, block, 0, stream>>>(
        f0, f1, f2, f3, indices, out);
}